// Dehazing_61641370632309
// MI455X (gfx1250) — compile-verified
//
#include <hip/hip_runtime.h>
#include <hip/hip_bf16.h>

typedef float v2f __attribute__((ext_vector_type(2)));
typedef float v8f __attribute__((ext_vector_type(8)));

#define NIMG   32
#define HW     (512 * 512)
#define WIMG   512
#define NBINS  1024
#define TOPK   13107          // int(512*512*0.05)

// ---------------- K1: per-pixel channel min ----------------
__global__ void k_minchan(const float* __restrict__ img, float* __restrict__ mn, int total) {
    int i = blockIdx.x * blockDim.x + threadIdx.x;
    if (i >= total) return;
    int n = i / HW, p = i % HW;
    const float* base = img + (size_t)n * 3 * HW + p;
    float a = base[0], b = base[HW], c = base[2 * HW];
    mn[i] = fminf(a, fminf(b, c));
}

// ---------------- K2: 15x15 box sum via WMMA banded matmul ----------------
// One wave (32 threads) per 16x16 output tile. S = R * M * R^T with
// R[i][j] = (|i-j| <= 7). Band < 16 so only the 3x3 tile neighborhood matters.
__global__ __launch_bounds__(32) void k_boxsum_wmma(const float* __restrict__ mn,
                                                    float* __restrict__ dc) {
    __shared__ float tiles[9][256];   // 3x3 input tiles, row-major [row*16+col]
    __shared__ float hbuf[3][256];    // horizontally-summed row-block tiles
    const int bx = blockIdx.x, by = blockIdx.y, n = blockIdx.z;
    const int l  = threadIdx.x;
    const int lo = l & 15;            // lane % 16
    const int hi = l >> 4;            // lane / 16 (0 or 1)
    const float* src = mn + (size_t)n * HW;

    // load 3x3 neighborhood of 16x16 tiles (zero-pad outside the image)
    for (int t = 0; t < 9; ++t) {
        int rb = t / 3 - 1, cb = t % 3 - 1;
        bool ok = ((unsigned)(by + rb) < 32u) && ((unsigned)(bx + cb) < 32u);
        int rbase = (by + rb) * 16, cbase = (bx + cb) * 16;
        for (int i = 0; i < 8; ++i) {
            int idx = l + 32 * i;
            int r = idx >> 4, c = idx & 15;
            tiles[t][idx] = ok ? src[(size_t)(rbase + r) * WIMG + (cbase + c)] : 0.0f;
        }
    }
    __syncthreads();

    // Horizontal pass: H_r = sum_db  M(r, bx+db) x B_db,  B_db[k][nn] = (|k+16*(db-1)-nn|<=7)
    for (int r = 0; r < 3; ++r) {
        v8f acc = {};
        for (int db = 0; db < 3; ++db) {
            const float* tp = tiles[r * 3 + db];
            for (int kk = 0; kk < 4; ++kk) {
                int k0 = 4 * kk + 2 * hi;            // K index held by this lane half
                v2f a, b;
                // A operand (data): a[v] = M[lane%16][k0+v]
                a.x = tp[lo * 16 + k0];
                a.y = tp[lo * 16 + k0 + 1];
                // B operand (banded weights): b[v] = B[k0+v][lane%16]
                int s0 = k0 + 16 * (db - 1) - lo;
                int s1 = s0 + 1;
                b.x = (s0 >= -7 && s0 <= 7) ? 1.0f : 0.0f;
                b.y = (s1 >= -7 && s1 <= 7) ? 1.0f : 0.0f;
                acc = __builtin_amdgcn_wmma_f32_16x16x4_f32(
                        false, a, false, b, (short)0, acc, false, false);
            }
        }
        // C layout: acc[v] = H[v + 8*hi][lo]
        for (int v = 0; v < 8; ++v) hbuf[r][(v + 8 * hi) * 16 + lo] = acc[v];
    }
    __syncthreads();

    // Vertical pass: S = sum_t  A_t x H_t,  A_t[m][k] = (|m-k-16*(t-1)|<=7)
    v8f out = {};
    for (int t = 0; t < 3; ++t) {
        const float* hp = hbuf[t];
        for (int kk = 0; kk < 4; ++kk) {
            int k0 = 4 * kk + 2 * hi;
            v2f a, b;
            int s0 = lo - k0 - 16 * (t - 1);
            int s1 = s0 - 1;
            a.x = (s0 >= -7 && s0 <= 7) ? 1.0f : 0.0f;
            a.y = (s1 >= -7 && s1 <= 7) ? 1.0f : 0.0f;
            b.x = hp[k0 * 16 + lo];
            b.y = hp[(k0 + 1) * 16 + lo];
            out = __builtin_amdgcn_wmma_f32_16x16x4_f32(
                    false, a, false, b, (short)0, out, false, false);
        }
    }
    float* dst = dc + (size_t)n * HW;
    const float inv225 = 1.0f / 225.0f;
    for (int v = 0; v < 8; ++v) {
        int row = by * 16 + v + 8 * hi;
        dst[(size_t)row * WIMG + bx * 16 + lo] = out[v] * inv225;
    }
}

// ---------------- K0: zero small accumulators ----------------
__global__ void k_zero(int* hist, float* sums, int* cnts) {
    int i = blockIdx.x * blockDim.x + threadIdx.x;
    if (i < NIMG * NBINS) hist[i] = 0;
    if (i < NIMG * 3) sums[i] = 0.0f;
    if (i < NIMG) cnts[i] = 0;
}

// ---------------- K3: per-image dc histogram ----------------
__global__ void k_hist(const float* __restrict__ dc, int* __restrict__ hist) {
    __shared__ int sh[NBINS];
    int n = blockIdx.y;
    for (int i = threadIdx.x; i < NBINS; i += blockDim.x) sh[i] = 0;
    __syncthreads();
    const float* src = dc + (size_t)n * HW;
    int base = blockIdx.x * (blockDim.x * 16);
    for (int i = 0; i < 16; ++i) {
        int p = base + threadIdx.x + i * blockDim.x;
        float v = src[p];
        int b = (int)(v * (float)NBINS);
        b = b < 0 ? 0 : (b > NBINS - 1 ? NBINS - 1 : b);
        atomicAdd(&sh[b], 1);
    }
    __syncthreads();
    int* gh = hist + n * NBINS;
    for (int i = threadIdx.x; i < NBINS; i += blockDim.x)
        if (sh[i]) atomicAdd(&gh[i], sh[i]);
}

// ---------------- K4: top-k threshold from histogram ----------------
__global__ void k_thresh(const int* __restrict__ hist, float* __restrict__ thresh) {
    int n = blockIdx.x * blockDim.x + threadIdx.x;
    if (n >= NIMG) return;
    const int* h = hist + n * NBINS;
    int cum = 0, b = NBINS - 1;
    for (; b >= 0; --b) { cum += h[b]; if (cum >= TOPK) break; }
    if (b < 0) b = 0;
    thresh[n] = (float)b / (float)NBINS;   // lower edge of threshold bin
}

// ---------------- K5: masked per-image channel sums ----------------
__global__ void k_accum(const float* __restrict__ img, const float* __restrict__ dc,
                        const float* __restrict__ thresh,
                        float* __restrict__ sums, int* __restrict__ cnts) {
    __shared__ float ssum[3];
    __shared__ int scnt;
    int n = blockIdx.y;
    if (threadIdx.x == 0) { ssum[0] = ssum[1] = ssum[2] = 0.0f; scnt = 0; }
    __syncthreads();
    float th = thresh[n];
    const float* d  = dc + (size_t)n * HW;
    const float* im = img + (size_t)n * 3 * HW;
    int base = blockIdx.x * (blockDim.x * 16);
    float l0 = 0.f, l1 = 0.f, l2 = 0.f; int lc = 0;
    for (int i = 0; i < 16; ++i) {
        int p = base + threadIdx.x + i * blockDim.x;
        if (d[p] >= th) {
            l0 += im[p]; l1 += im[p + HW]; l2 += im[p + 2 * HW]; ++lc;
        }
    }
    if (lc) {
        atomicAdd(&ssum[0], l0); atomicAdd(&ssum[1], l1);
        atomicAdd(&ssum[2], l2); atomicAdd(&scnt, lc);
    }
    __syncthreads();
    if (threadIdx.x == 0 && scnt) {
        atomicAdd(&sums[n * 3 + 0], ssum[0]);
        atomicAdd(&sums[n * 3 + 1], ssum[1]);
        atomicAdd(&sums[n * 3 + 2], ssum[2]);
        atomicAdd(&cnts[n], scnt);
    }
}

// ---------------- K6: A = sum / count ----------------
__global__ void k_alight(const float* __restrict__ sums, const int* __restrict__ cnts,
                         float* __restrict__ A) {
    int i = threadIdx.x;
    if (i >= NIMG * 3) return;
    int c = cnts[i / 3]; if (c < 1) c = 1;
    A[i] = sums[i] / (float)c;
}

// ---------------- K7: fused transmission + recovery + clip ----------------
__global__ void k_final(const float* __restrict__ img, const float* __restrict__ dc,
                        const float* __restrict__ A, const float* __restrict__ w,
                        float* __restrict__ out, int total) {
    int i = blockIdx.x * blockDim.x + threadIdx.x;
    if (i >= total) return;
    int n = i / HW, p = i % HW;
    float t = 1.0f - w[n] * dc[i];
    t = t > 0.1f ? t : 0.1f;
    float inv = 1.0f / (t + 0.001f);
    const float* im = img + (size_t)n * 3 * HW + p;
    float* o = out + (size_t)n * 3 * HW + p;
    for (int c = 0; c < 3; ++c) {
        float a = A[n * 3 + c];
        float x = (im[(size_t)c * HW] - a) * inv + a;
        x = x < 0.0f ? 0.0f : (x > 1.0f ? 1.0f : x);
        o[(size_t)c * HW] = x;
    }
}

extern "C" void kernel_launch(void* const* d_in, const int* in_sizes, int n_in,
                              void* d_out, int out_size, void* d_ws, size_t ws_size,
                              hipStream_t stream) {
    const float* img = (const float*)d_in[0];   // (32,3,512,512) fp32
    const float* w   = (const float*)d_in[1];   // (32,) fp32
    float* out = (float*)d_out;

    char* ws = (char*)d_ws;
    const size_t planeBytes = (size_t)NIMG * HW * sizeof(float);   // 33.5 MB
    float* dc     = (float*)ws;
    float* mn     = (float*)(ws + planeBytes);
    int*   hist   = (int*)(ws + 2 * planeBytes);
    float* thresh = (float*)(ws + 2 * planeBytes + (size_t)NIMG * NBINS * sizeof(int));
    float* sums   = thresh + NIMG;
    int*   cnts   = (int*)(sums + NIMG * 3);
    float* A      = (float*)(cnts + NIMG);

    const int totalPix = NIMG * HW;

    k_zero<<<(NIMG * NBINS + 255) / 256, 256, 0, stream>>>(hist, sums, cnts);
    k_minchan<<<(totalPix + 255) / 256, 256, 0, stream>>>(img, mn, totalPix);

    dim3 g2(32, 32, 32);                 // (bx, by, n)
    k_boxsum_wmma<<<g2, 32, 0, stream>>>(mn, dc);

    dim3 g3(64, NIMG);                   // 64 blocks * 256 thr * 16 px = 262144
    k_hist<<<g3, 256, 0, stream>>>(dc, hist);
    k_thresh<<<1, 32, 0, stream>>>(hist, thresh);
    k_accum<<<g3, 256, 0, stream>>>(img, dc, thresh, sums, cnts);
    k_alight<<<1, 96, 0, stream>>>(sums, cnts, A);
    k_final<<<(totalPix + 255) / 256, 256, 0, stream>>>(img, dc, A, w, out, totalPix);
}